// DiversificationBlock_21431886807355
// MI455X (gfx1250) — compile-verified
//
#include <hip/hip_runtime.h>
#include <stdint.h>

// Problem shape (fixed by setup_inputs): x[16,200,112,112] fp32, seed int.
#define B_      16
#define C_      200
#define H_      112
#define W_      112
#define G_      7
#define NH_     16              // H/G
#define NW_     16              // W/G
#define MAP_    (H_*W_)         // 12544 elements per (b,c) map
#define NMAPS_  (B_*C_)         // 3200 maps -> 3200 workgroups
#define NPATCH_ (NH_*NW_)       // 256 patches -> 1 per thread
#define THREADS_ 256
#define PEAK_HALF_  ((unsigned)(NMAPS_/2))            // 1600
#define PATCH_HALF_ ((unsigned)((NMAPS_*NPATCH_)/2))  // 409600
#define ALPHA_  0.1f

typedef unsigned int u32;
typedef u32 u32x4 __attribute__((ext_vector_type(4)));
typedef int  i32x4 __attribute__((ext_vector_type(4)));
typedef int  i32x8 __attribute__((ext_vector_type(8)));

// ---------- CDNA5 TDM path selection (compile-time probes) ----------
#if defined(__gfx1250__) && __has_builtin(__builtin_amdgcn_tensor_load_to_lds)
#define TDM_MODE 1
#else
#define TDM_MODE 0
#endif

__device__ __forceinline__ void wait_tensorcnt0() {
#if defined(__gfx1250__)
#if __has_builtin(__builtin_amdgcn_s_wait_tensorcnt)
  __builtin_amdgcn_s_wait_tensorcnt(0);
#else
  asm volatile("s_wait_tensorcnt 0x0" ::: "memory");
#endif
#endif
}

#if TDM_MODE
// D# group 0: count=1 | lds_addr | global_addr[56:0] | type=2 (see ISA 08 §8.3)
__device__ __forceinline__ u32x4 tdm_g0(u32 lds_byte_addr, unsigned long long gaddr) {
  u32x4 g;
  g.x = 1u;                                            // count=1, user mode
  g.y = lds_byte_addr;                                 // LDS byte offset
  g.z = (u32)(gaddr & 0xFFFFFFFFull);                  // global_addr[31:0]
  g.w = (u32)((gaddr >> 32) & 0x1FFFFFFull) | (2u << 30); // [56:32] | type=2
  return g;
}
// D# group 1: 1-D tile of MAP_ fp32 elements (data_size=4B). (ISA 08 §8.4)
__device__ __forceinline__ i32x8 tdm_g1() {
  i32x8 g;
  g[0] = (int)(2u << 16);            // wg_mask=0, data_size=2 (4 bytes)
  g[1] = (int)((u32)MAP_ << 16);     // tensor_dim0[15:0] = 12544
  g[2] = (int)(1u << 16);            // tensor_dim1 = 1
  g[3] = (int)((u32)MAP_ << 16);     // tile_dim0 = 12544
  g[4] = 1;                          // tile_dim1 = 1, tile_dim2 = 0
  g[5] = MAP_;                       // tensor_dim0_stride[31:0]
  g[6] = 0;                          // stride hi / dim1_stride lo
  g[7] = 0;                          // dim1_stride hi
  return g;
}
// amdgpu-toolchain (clang-23) arity: 6 args (g0, g1, g2, g3, g4, cpol)
__device__ __forceinline__ void tdm_load(u32x4 g0, i32x8 g1) {
  i32x4 z4 = {0, 0, 0, 0};
  i32x8 z8 = {0, 0, 0, 0, 0, 0, 0, 0};
  __builtin_amdgcn_tensor_load_to_lds(g0, g1, z4, z4, z8, 0);
}
__device__ __forceinline__ void tdm_store(u32x4 g0, i32x8 g1) {
  i32x4 z4 = {0, 0, 0, 0};
  i32x8 z8 = {0, 0, 0, 0, 0, 0, 0, 0};
  __builtin_amdgcn_tensor_store_from_lds(g0, g1, z4, z4, z8, 0);
}
#endif // TDM_MODE

// ---------- Exact JAX Threefry-2x32 ----------
__device__ __forceinline__ u32 rotl32(u32 x, int n) { return (x << n) | (x >> (32 - n)); }

__device__ __forceinline__ void threefry2x32(u32 k0, u32 k1, u32 c0, u32 c1,
                                             u32 &o0, u32 &o1) {
  const u32 ks0 = k0, ks1 = k1, ks2 = k0 ^ k1 ^ 0x1BD11BDAu;
  u32 x0 = c0 + ks0, x1 = c1 + ks1;
#define TF_R(r) { x0 += x1; x1 = rotl32(x1, (r)); x1 ^= x0; }
  TF_R(13) TF_R(15) TF_R(26) TF_R(6)   x0 += ks1; x1 += ks2 + 1u;
  TF_R(17) TF_R(29) TF_R(16) TF_R(24)  x0 += ks2; x1 += ks0 + 2u;
  TF_R(13) TF_R(15) TF_R(26) TF_R(6)   x0 += ks0; x1 += ks1 + 3u;
  TF_R(17) TF_R(29) TF_R(16) TF_R(24)  x0 += ks1; x1 += ks2 + 4u;
  TF_R(13) TF_R(15) TF_R(26) TF_R(6)   x0 += ks2; x1 += ks0 + 5u;
#undef TF_R
  o0 = x0; o1 = x1;
}

// element i of jax random_bits(key,32,(n,)) with n even, half = n/2
__device__ __forceinline__ u32 tf_bits(u32 k0, u32 k1, u32 i, u32 half) {
  u32 a, b;
  if (i < half) { threefry2x32(k0, k1, i, i + half, a, b); return a; }
  threefry2x32(k0, k1, i - half, i, a, b); return b;
}

// ---------- Kernel: one workgroup per (b,c) map ----------
__global__ __launch_bounds__(THREADS_)
void divblock_kernel(const float* __restrict__ x, const int* __restrict__ seed_p,
                     float* __restrict__ out) {
  __shared__ float tile[MAP_];            // 50176 B: whole map staged in LDS
  __shared__ float red[THREADS_];
  __shared__ unsigned char pmflag[NPATCH_];

  const int t = threadIdx.x;
  const int m = blockIdx.x;               // flat (b*C + c) map index

  // -------- stage map into LDS (TDM async copy, single DMA per block) -----
#if TDM_MODE
  const u32 lds_tile = (u32)(uintptr_t)(void*)tile;   // flat addr[31:0] == LDS offset
  if (t < 32) {                           // wave 0 only (EXEC all-ones in wave)
    tdm_load(tdm_g0(lds_tile,
                    (unsigned long long)(uintptr_t)(x + (size_t)m * MAP_)),
             tdm_g1());
    wait_tensorcnt0();
  }
  __syncthreads();
#else
  {
    const float* src = x + (size_t)m * MAP_;
    for (int i = t; i < MAP_; i += THREADS_) tile[i] = src[i];
    __syncthreads();
  }
#endif

  // -------- derive k_peak / k_patch exactly like jax.random.split --------
  const u32 seed = (u32)seed_p[0];
  u32 a0, a1, b0, b1;
  threefry2x32(0u, seed, 0u, 2u, a0, b0);   // lane 0 of iota(4) halves
  threefry2x32(0u, seed, 1u, 3u, a1, b1);   // lane 1
  // k_peak = (a0,a1), k_patch = (b0,b1)

  // -------- per-patch max (1 patch per thread; stride-7 is conflict-free) -
  const int ph = t >> 4, pw = t & 15;
  const float* prow = &tile[(ph * G_) * W_ + pw * G_];
  float pmax = prow[0];
#pragma unroll
  for (int r = 0; r < G_; ++r)
#pragma unroll
    for (int c = 0; c < G_; ++c)
      pmax = fmaxf(pmax, prow[r * W_ + c]);

  red[t] = pmax;
  __syncthreads();
  for (int s = THREADS_ / 2; s > 0; s >>= 1) {
    if (t < s) red[t] = fmaxf(red[t], red[t + s]);
    __syncthreads();
  }
  const float maxv = red[0];               // map max == max of patch maxes

  // -------- Bernoulli bits: u<0.5  <=>  top bit of random word is 0 ------
  const bool rp     = (tf_bits(a0, a1, (u32)m, PEAK_HALF_) & 0x80000000u) == 0u;
  const u32  j      = (u32)m * NPATCH_ + (u32)t;
  const bool rpatch = (tf_bits(b0, b1, j, PATCH_HALF_) & 0x80000000u) == 0u;

  pmflag[t] = (rpatch && (pmax != maxv)) ? 1 : 0;   // r_patch & ~has_max
  __syncthreads();

  // -------- apply suppression in LDS (m in {0,1}; overlap impossible) ----
  for (int i = t; i < MAP_; i += THREADS_) {
    const float v = tile[i];
    const int h = i / W_, w = i - h * W_;
    const int p = (h / G_) * NW_ + (w / G_);
    const bool msk = pmflag[p] || (rp && (v == maxv));
    tile[i] = msk ? v * ALPHA_ : v;
  }
  __syncthreads();

  // -------- write map back (TDM async store from LDS) --------------------
#if TDM_MODE
  if (t < 32) {
    tdm_store(tdm_g0(lds_tile,
                     (unsigned long long)(uintptr_t)(out + (size_t)m * MAP_)),
              tdm_g1());
    wait_tensorcnt0();   // S_ENDPGM also implies wait-idle, belt & suspenders
  }
#else
  {
    float* dst = out + (size_t)m * MAP_;
    for (int i = t; i < MAP_; i += THREADS_) dst[i] = tile[i];
  }
#endif
}

extern "C" void kernel_launch(void* const* d_in, const int* in_sizes, int n_in,
                              void* d_out, int out_size, void* d_ws, size_t ws_size,
                              hipStream_t stream) {
  const float* x    = (const float*)d_in[0];
  const int*   seed = (const int*)d_in[1];
  float*       out  = (float*)d_out;
  (void)in_sizes; (void)n_in; (void)out_size; (void)d_ws; (void)ws_size;
  divblock_kernel<<<dim3(NMAPS_), dim3(THREADS_), 0, stream>>>(x, seed, out);
}